// Kernel_39814346834171
// MI455X (gfx1250) — compile-verified
//
#include <hip/hip_runtime.h>

// ---------------------------------------------------------------------------
// Fused equivariant-filter kernel for MI455X (gfx1250, wave32, WMMA).
//   out[z, i*18+j] = sum_k cg[k,i,j] * (Rm[z,k] * Ym[z,k])
// One workgroup (256 thr = 8 waves) handles ZT=128 z-points entirely in LDS.
// All GEMMs use v_wmma_f32_16x16x4_f32 (exact fp32; op is HBM-bound anyway).
// ---------------------------------------------------------------------------

typedef __attribute__((ext_vector_type(2))) float v2f;
typedef __attribute__((ext_vector_type(8))) float v8f;

#define ZT    128
#define KDIM  204   // total filter components (multiple of 4 -> clean K loop)
#define NPATH 60
#define IJ    324   // 18*18 output components per z
#define NCT   21    // ceil(324/16) column tiles

// LDS offsets in 4-byte words (total 64544 words = 258176 B < 320 KB/WGP)
#define OFF_W1   0        // 64 x 68  (stride-padded W1)
#define OFF_W2   4352     // 64 x 68  (W2 zero-padded 60 -> 64 cols)
#define OFF_B1   8704     // 64
#define OFF_B2   8768     // 64 (zero-padded)
#define OFF_PATH 8832     // 208 ints: path index per k
#define OFF_YIDX 9040     // 208 ints: ylm index per k
#define OFF_Y    9248     // 128 x 16 spherical harmonics
#define OFF_B    11296    // 128 x 68 radial basis
#define OFF_H    20000    // 128 x 68 hidden
#define OFF_R    28704    // 128 x 68 MLP output (cols 60..63 = 0)
#define OFF_W    37408    // 128 x 212 expanded weights w[z,k]
#define LDS_WORDS 64544

__global__ void __launch_bounds__(256) eq_filter_fused(
    const float* __restrict__ gR,  const float* __restrict__ gW1,
    const float* __restrict__ gb1, const float* __restrict__ gW2,
    const float* __restrict__ gb2, const float* __restrict__ gCG,
    float* __restrict__ gOut, int Z)
{
    extern __shared__ float sm[];
    float* sW1 = sm + OFF_W1;
    float* sW2 = sm + OFF_W2;
    float* sb1 = sm + OFF_B1;
    float* sb2 = sm + OFF_B2;
    int*   sPath = (int*)(sm + OFF_PATH);
    int*   sYidx = (int*)(sm + OFF_YIDX);
    float* sY  = sm + OFF_Y;
    float* sB  = sm + OFF_B;
    float* sH  = sm + OFF_H;
    float* sRm = sm + OFF_R;
    float* sWm = sm + OFF_W;

    const int tid = threadIdx.x;
    const int z0  = blockIdx.x * ZT;

    // ---------------- phase 0a: stage weights, build k -> (path, ylm) table
    for (int idx = tid; idx < 64 * 64; idx += 256) {
        int h = idx >> 6, c = idx & 63;
        sW1[h * 68 + c] = gW1[idx];
        sW2[h * 68 + c] = (c < NPATH) ? gW2[h * NPATH + c] : 0.0f;
    }
    if (tid < 64) {
        sb1[tid] = gb1[tid];
        sb2[tid] = (tid < NPATH) ? gb2[tid] : 0.0f;
    }
    if (tid == 0) {   // replicate reference _filter_paths() / mix structure
        const int yoff[3] = {0, 1, 4};
        int k = 0, p = 0;
        for (int li = 0; li < 3; ++li)
            for (int lo = 0; lo < 3; ++lo) {
                int lmin = (li > lo) ? li - lo : lo - li;
                int lmax = li + lo; if (lmax > 2) lmax = 2;
                for (int lf = lmin; lf <= lmax; ++lf)
                    for (int rep = 0; rep < 4; ++rep) {   // MUL*MUL
                        for (int m = 0; m < 2 * lf + 1; ++m) {
                            sPath[k] = p; sYidx[k] = yoff[lf] + m; ++k;
                        }
                        ++p;
                    }
            }
    }

    // ---------------- phase 0b: radial basis + spherical harmonics
    {
        const int zloc = tid & 127;
        const int half = tid >> 7;
        int gz = z0 + zloc; if (gz >= Z) gz = Z - 1;
        float x = gR[gz * 3 + 0], y = gR[gz * 3 + 1], zc = gR[gz * 3 + 2];
        float rad = sqrtf(x * x + y * y + zc * zc);
        float inv = 1.0f / (rad + 1e-12f);
        x *= inv; y *= inv; zc *= inv;
        if (half == 0) {
            const float c0 = 0.28209479177387814f, c1 = 0.4886025119029199f;
            const float c2a = 1.0925484305920792f, c2b = 0.31539156525252005f;
            const float c2c = 0.5462742152960396f;
            float* Yr = sY + zloc * 16;
            Yr[0] = c0;          Yr[1] = c1 * y;      Yr[2] = c1 * zc;
            Yr[3] = c1 * x;      Yr[4] = c2a * x * y; Yr[5] = c2a * y * zc;
            Yr[6] = c2b * (3.0f * zc * zc - 1.0f);
            Yr[7] = c2a * x * zc; Yr[8] = c2c * (x * x - y * y);
        }
        const float step = 3.5f / 63.0f;   // linspace(0, R_MAX, 64)
        for (int j = half * 32; j < half * 32 + 32; ++j) {
            float d = rad - step * (float)j;
            sB[zloc * 68 + j] = __expf(-4.0f * d * d);   // GAMMA = 4
        }
    }
    __syncthreads();

    const int wave = tid >> 5;
    const int lane = tid & 31;
    const int M    = lane & 15;          // A-matrix row / B,C,D column
    const int kb   = (lane >> 4) << 1;   // k-pair base within 4-wide K step
    const int hiM  = (lane >> 4) << 3;   // C/D row offset (rows v / v+8)

    // ---------------- phase 1: H = relu(B @ W1 + b1)   (wave: z-tile = wave)
    {
        const int zt = wave;
        for (int ht = 0; ht < 4; ++ht) {
            v8f acc = {0.f, 0.f, 0.f, 0.f, 0.f, 0.f, 0.f, 0.f};
#pragma unroll
            for (int k0 = 0; k0 < 64; k0 += 4) {
                v2f a, b;
                a.x = sB[(zt * 16 + M) * 68 + k0 + kb];
                a.y = sB[(zt * 16 + M) * 68 + k0 + kb + 1];
                b.x = sW1[(k0 + kb) * 68 + ht * 16 + M];
                b.y = sW1[(k0 + kb + 1) * 68 + ht * 16 + M];
                acc = __builtin_amdgcn_wmma_f32_16x16x4_f32(
                    false, a, false, b, (short)0, acc, false, false);
            }
            const int col = ht * 16 + M;
            const float bias = sb1[col];
#pragma unroll
            for (int v = 0; v < 8; ++v) {
                float hv = acc[v] + bias;
                sH[(zt * 16 + hiM + v) * 68 + col] = hv > 0.0f ? hv : 0.0f;
            }
        }
    }
    __syncthreads();

    // ---------------- phase 2: Rmlp = H @ W2 + b2
    {
        const int zt = wave;
        for (int ht = 0; ht < 4; ++ht) {
            v8f acc = {0.f, 0.f, 0.f, 0.f, 0.f, 0.f, 0.f, 0.f};
#pragma unroll
            for (int k0 = 0; k0 < 64; k0 += 4) {
                v2f a, b;
                a.x = sH[(zt * 16 + M) * 68 + k0 + kb];
                a.y = sH[(zt * 16 + M) * 68 + k0 + kb + 1];
                b.x = sW2[(k0 + kb) * 68 + ht * 16 + M];
                b.y = sW2[(k0 + kb + 1) * 68 + ht * 16 + M];
                acc = __builtin_amdgcn_wmma_f32_16x16x4_f32(
                    false, a, false, b, (short)0, acc, false, false);
            }
            const int col = ht * 16 + M;
            const float bias = sb2[col];
#pragma unroll
            for (int v = 0; v < 8; ++v)
                sRm[(zt * 16 + hiM + v) * 68 + col] = acc[v] + bias;
        }
    }
    __syncthreads();

    // ---------------- phase 3: expand w[z,k] = R[z, path(k)] * Y[ylm(k), z]
    {
        const int zloc = tid & 127;
        const int half = tid >> 7;
        for (int kk = half * 104; kk < half * 104 + 104; ++kk) {
            float wv = 0.0f;
            if (kk < KDIM)
                wv = sRm[zloc * 68 + sPath[kk]] * sY[zloc * 16 + sYidx[kk]];
            sWm[zloc * 212 + kk] = wv;
        }
    }
    __syncthreads();

    // ---------------- phase 4: out[z, 0:324] = w[z, 0:204] @ cg[204, 324]
    // Wave owns column tiles; cg B-fragments loaded from global (L2-resident)
    // ONCE per k-step and reused across 8 z-tile accumulators.
    for (int ct = wave; ct < NCT; ct += 8) {
        const int col = ct * 16 + M;
        const bool colOK = (col < IJ);
        v8f acc[8];
#pragma unroll
        for (int zt = 0; zt < 8; ++zt)
            acc[zt] = (v8f){0.f, 0.f, 0.f, 0.f, 0.f, 0.f, 0.f, 0.f};
#pragma unroll 1
        for (int k0 = 0; k0 < KDIM; k0 += 4) {
            v2f b;
            b.x = colOK ? gCG[(k0 + kb) * IJ + col] : 0.0f;
            b.y = colOK ? gCG[(k0 + kb + 1) * IJ + col] : 0.0f;
#pragma unroll
            for (int zt = 0; zt < 8; ++zt) {
                v2f a;
                a.x = sWm[(zt * 16 + M) * 212 + k0 + kb];
                a.y = sWm[(zt * 16 + M) * 212 + k0 + kb + 1];
                acc[zt] = __builtin_amdgcn_wmma_f32_16x16x4_f32(
                    false, a, false, b, (short)0, acc[zt], false, false);
            }
        }
#pragma unroll
        for (int zt = 0; zt < 8; ++zt) {
#pragma unroll
            for (int v = 0; v < 8; ++v) {
                const int gz = z0 + zt * 16 + hiM + v;
                if (colOK && gz < Z)
                    gOut[(size_t)gz * IJ + col] = acc[zt][v];
            }
        }
    }
}

extern "C" void kernel_launch(void* const* d_in, const int* in_sizes, int n_in,
                              void* d_out, int out_size, void* d_ws, size_t ws_size,
                              hipStream_t stream) {
    const float* r   = (const float*)d_in[0];
    const float* W1  = (const float*)d_in[1];
    const float* b1  = (const float*)d_in[2];
    const float* W2  = (const float*)d_in[3];
    const float* b2  = (const float*)d_in[4];
    const float* cg  = (const float*)d_in[5];
    // d_in[6] (rf_mix) / d_in[7] (ylm_mix) are 0/1 selectors; the index table
    // is regenerated on-device from the path structure instead.
    float* out = (float*)d_out;
    const int Z = in_sizes[0] / 3;

    const int grid = (Z + ZT - 1) / ZT;
    const size_t lds_bytes = (size_t)LDS_WORDS * sizeof(float);
    eq_filter_fused<<<grid, 256, lds_bytes, stream>>>(r, W1, b1, W2, b2, cg, out, Z);
}